// AttnDecoderRNN_16836271800433
// MI455X (gfx1250) — compile-verified
//
#include <hip/hip_runtime.h>
#include <math.h>

typedef __attribute__((ext_vector_type(2))) float v2f;
typedef __attribute__((ext_vector_type(8))) float v8f;

#define HIDDEN 1024
#define VOCAB  50257
#define MAXLEN 64

// ---------------------------------------------------------------------------
// Generic 16-rows-per-wave fp32 matvec using V_WMMA_F32_16X16X4_F32.
//   out[r] = act( sum_k W[r,k]*x[k] + bias[r] ),  W row-major [nrows x K]
// A tile = W rows (lane&15 = row, lane>>4 selects K pair) -> global b64 loads
// B tile = x segment in column 0 only (lanes 0 and 16), zeros elsewhere
// C/D   = accumulator; column 0 read back from lanes 0 / 16.
// K <= 2048, staged in LDS once per block. 8 waves (= 8 row tiles) per block.
// ---------------------------------------------------------------------------
__global__ void __launch_bounds__(256)
matvec16_wmma(const float* __restrict__ W, const float* __restrict__ x,
              const float* __restrict__ bias, float* __restrict__ out,
              int nrows, int K, int act_relu)
{
    __shared__ float xs[2048];
    for (int i = threadIdx.x; i < K; i += 256) xs[i] = x[i];
    __syncthreads();

    const int lane    = threadIdx.x & 31;
    const int wave    = threadIdx.x >> 5;
    const int tile    = blockIdx.x * 8 + wave;
    const int m0      = tile * 16;
    if (m0 >= nrows) return;                 // wave-uniform: EXEC stays all-1s

    const int halfsel = lane >> 4;           // 0: K={0,1}, 1: K={2,3}
    const int rlane   = lane & 15;
    int row = m0 + rlane;
    if (row >= nrows) row = nrows - 1;       // clamp (stores are guarded)
    const float* wrow = W + (size_t)row * (size_t)K + (size_t)(halfsel << 1);

    v8f acc = {0.f, 0.f, 0.f, 0.f, 0.f, 0.f, 0.f, 0.f};

#pragma unroll 4
    for (int k0 = 0; k0 < K; k0 += 4) {
        v2f a = *(const v2f*)(wrow + k0);                       // W[row, k0+2h .. +1]
        v2f bx = *(const v2f*)(xs + k0 + (halfsel << 1));       // broadcast LDS read
        v2f b;
        b.x = (rlane == 0) ? bx.x : 0.f;                        // column N=0 only
        b.y = (rlane == 0) ? bx.y : 0.f;
        acc = __builtin_amdgcn_wmma_f32_16x16x4_f32(
                  false, a, false, b, (short)0, acc, false, false);
    }

    // Column 0 of D: lane 0 -> rows m0..m0+7, lane 16 -> rows m0+8..m0+15
    if (rlane == 0) {
        const int base = m0 + (halfsel << 3);
#pragma unroll
        for (int i = 0; i < 8; ++i) {
            const int r = base + i;
            if (r < nrows) {
                float v = acc[i] + bias[r];
                if (act_relu) v = fmaxf(v, 0.f);
                out[r] = v;
            }
        }
    }
}

// ---------------------------------------------------------------------------
// Attention prologue (single block): embedding gather, 64 attention scores,
// softmax, context vector; emits u = [embedded, context] (2048) and weights.
// ---------------------------------------------------------------------------
__global__ void __launch_bounds__(256)
attn_prologue(const int* __restrict__ token, const float* __restrict__ hidden,
              const float* __restrict__ enc, const float* __restrict__ emb,
              const float* __restrict__ attn_W, const float* __restrict__ attn_b,
              float* __restrict__ u, float* __restrict__ attn_w_out)
{
    __shared__ float cat[2 * HIDDEN];
    __shared__ float sc[MAXLEN];
    __shared__ float wts[MAXLEN];

    const int t    = threadIdx.x;
    const int lane = t & 31;
    const int wave = t >> 5;

    const int tok = token[0];                       // int64 LE: low word is value
    const float* erow = emb + (size_t)tok * HIDDEN;
    for (int i = t; i < HIDDEN; i += 256) {
        cat[i]          = erow[i];
        cat[HIDDEN + i] = hidden[i];
    }
    __syncthreads();

    // scores: 8 waves x 8 rows, lane-strided dot over 2048 + shuffle reduce
    for (int r8 = 0; r8 < 8; ++r8) {
        const int r = wave * 8 + r8;
        const float* wr = attn_W + (size_t)r * (2 * HIDDEN);
        float p = 0.f;
        for (int j = lane; j < 2 * HIDDEN; j += 32) p += wr[j] * cat[j];
        for (int off = 16; off; off >>= 1) p += __shfl_xor(p, off, 32);
        if (lane == 0) sc[r] = p + attn_b[r];
    }
    __syncthreads();

    // softmax over 64 by wave 0 (2 values per lane)
    if (wave == 0) {
        float a = sc[lane], b = sc[lane + 32];
        float m = fmaxf(a, b);
        for (int off = 16; off; off >>= 1) m = fmaxf(m, __shfl_xor(m, off, 32));
        float ea = expf(a - m), eb = expf(b - m);
        float s = ea + eb;
        for (int off = 16; off; off >>= 1) s += __shfl_xor(s, off, 32);
        const float inv = 1.f / s;
        wts[lane] = ea * inv;  wts[lane + 32] = eb * inv;
        attn_w_out[lane] = ea * inv;  attn_w_out[lane + 32] = eb * inv;
    }
    __syncthreads();

    // context = wts @ enc, and emit u = [embedded, context]
    for (int h = t; h < HIDDEN; h += 256) {
        float acc = 0.f;
        for (int i = 0; i < MAXLEN; ++i) acc += wts[i] * enc[i * HIDDEN + h];
        u[HIDDEN + h] = acc;
        u[h]          = cat[h];
    }
}

// ---------------------------------------------------------------------------
// GRU gate math (PyTorch convention), 1024 hidden units.
// ---------------------------------------------------------------------------
__global__ void __launch_bounds__(256)
gru_gate(const float* __restrict__ gi, const float* __restrict__ gh,
         const float* __restrict__ h0, float* __restrict__ hnew)
{
    const int j = blockIdx.x * 256 + threadIdx.x;
    if (j < HIDDEN) {
        const float r = 1.f / (1.f + expf(-(gi[j]            + gh[j])));
        const float z = 1.f / (1.f + expf(-(gi[j + HIDDEN]   + gh[j + HIDDEN])));
        const float n = tanhf(gi[j + 2 * HIDDEN] + r * gh[j + 2 * HIDDEN]);
        hnew[j] = (1.f - z) * n + z * h0[j];
    }
}

// ---------------------------------------------------------------------------
extern "C" void kernel_launch(void* const* d_in, const int* in_sizes, int n_in,
                              void* d_out, int out_size, void* d_ws, size_t ws_size,
                              hipStream_t stream)
{
    const int*   token   = (const int*)  d_in[0];
    const float* hidden  = (const float*)d_in[1];
    const float* enc     = (const float*)d_in[2];
    const float* emb     = (const float*)d_in[3];
    const float* attn_W  = (const float*)d_in[4];
    const float* attn_b  = (const float*)d_in[5];
    const float* comb_W  = (const float*)d_in[6];
    const float* comb_b  = (const float*)d_in[7];
    const float* W_ih    = (const float*)d_in[8];
    const float* W_hh    = (const float*)d_in[9];
    const float* b_ih    = (const float*)d_in[10];
    const float* b_hh    = (const float*)d_in[11];
    const float* out_W   = (const float*)d_in[12];
    const float* out_b   = (const float*)d_in[13];

    float* out    = (float*)d_out;
    float* logits = out;                    // [50257]
    float* hnew   = out + VOCAB;            // [1024]
    float* attnw  = out + VOCAB + HIDDEN;   // [64]

    float* ws = (float*)d_ws;
    float* u  = ws;                         // 2048: [embedded, context]
    float* x  = ws + 2048;                  // 1024: relu(comb)
    float* gi = ws + 3072;                  // 3072
    float* gh = ws + 6144;                  // 3072

    auto nblocks = [](int nrows) { int tiles = (nrows + 15) / 16; return (tiles + 7) / 8; };

    attn_prologue<<<1, 256, 0, stream>>>(token, hidden, enc, emb, attn_W, attn_b, u, attnw);
    matvec16_wmma<<<nblocks(HIDDEN),     256, 0, stream>>>(comb_W, u,      comb_b, x,      HIDDEN,     2 * HIDDEN, 1);
    matvec16_wmma<<<nblocks(3 * HIDDEN), 256, 0, stream>>>(W_ih,   x,      b_ih,   gi,     3 * HIDDEN, HIDDEN,     0);
    matvec16_wmma<<<nblocks(3 * HIDDEN), 256, 0, stream>>>(W_hh,   hidden, b_hh,   gh,     3 * HIDDEN, HIDDEN,     0);
    gru_gate<<<(HIDDEN + 255) / 256, 256, 0, stream>>>(gi, gh, hidden, hnew);
    matvec16_wmma<<<nblocks(VOCAB),      256, 0, stream>>>(out_W,  hnew,   out_b,  logits, VOCAB,      HIDDEN,     0);
}